// AttentionBlock_9766755631474
// MI455X (gfx1250) — compile-verified
//
#include <hip/hip_runtime.h>

// MI455X (gfx1250) wave32 WMMA implementation.
// Stages:
//   1) cx = X_flat[6272,1024]  @ Wx^T -> ws  (padded to [6272,64])
//   2) ch = h_flat[16384,1024] @ Wh^T -> ws  (padded to [16384,64])
//   3) per (b, t-tile): z = sum_r tanh(cx+ch)*Wa, softmax over K,
//      out = alpha @ X[b]   (13 K-steps of wmma per 16-col tile)
// All GEMM work uses v_wmma_f32_16x16x4_f32 (f32 in/out, no precision loss).
//
// Design notes:
//  * Guard-free hot loops: padded-row indices are CLAMPED to valid memory
//    (no EXEC-mask branches). Safe because proj's padded output columns are
//    never read, and attn's alpha is exactly 0 for k>=49.
//  * Single-wave (32-thread) workgroups: no cross-wave LDS sharing anywhere,
//    so 1-wave blocks maximize WGP load balancing and need no barriers
//    (same-wave DS ops are in-order per the ISA).
//  * tanh: native V_TANH_F32 via __builtin_amdgcn_tanhf when available
//    (39M tanh evals otherwise rival the 160MB memory roofline in cost).

typedef __attribute__((ext_vector_type(2))) float v2f;
typedef __attribute__((ext_vector_type(8))) float v8f;

#define HDIM 1024
#define BDIM 128
#define TDIM 128
#define KREG 49     // regions (K axis) == projection dim (r axis) == 49
#define RPAD 64     // padded r dimension in workspace

__device__ __forceinline__ int imin(int a, int b) { return a < b ? a : b; }

#if __has_builtin(__builtin_amdgcn_tanhf)
__device__ __forceinline__ float fast_tanh(float x) { return __builtin_amdgcn_tanhf(x); }
#else
__device__ __forceinline__ float fast_tanh(float x) { return tanhf(x); }
#endif

// ---------------------------------------------------------------------------
// Stage 1/2: C[M][64] = A[M][1024] @ W[49][1024]^T
// One wave (= one 32-thread block) per 16-row M tile; computes a 16x64 tile
// (4 accumulators), reusing each A fragment for 4 WMMAs.
// ---------------------------------------------------------------------------
__global__ __launch_bounds__(32)
void proj_kernel(const float* __restrict__ A, const float* __restrict__ W,
                 float* __restrict__ C) {
  const int lane  = threadIdx.x & 31;
  const int mtile = blockIdx.x;

  const int row15 = lane & 15;        // M row (A) / N col (B) within tile
  const int koff  = (lane >> 4) * 2;  // K pair: lanes<16 -> {0,1}, else {2,3}

  const long arow = (long)mtile * 16 + row15;
  const float* Aptr = A + arow * HDIM + koff;

  // B-matrix rows are W rows (output column index = r); clamp padded rows:
  // their products land in output columns 49..63 which are never consumed.
  const float* W0 = W + (long)(row15 +  0)               * HDIM + koff;
  const float* W1 = W + (long)(row15 + 16)               * HDIM + koff;
  const float* W2 = W + (long)(row15 + 32)               * HDIM + koff;
  const float* W3 = W + (long)imin(row15 + 48, KREG - 1) * HDIM + koff;

  v8f acc0 = {}, acc1 = {}, acc2 = {}, acc3 = {};

  for (int h = 0; h < HDIM; h += 4) {
    v2f a  = *(const v2f*)(Aptr + h);
    v2f b0 = *(const v2f*)(W0 + h);
    v2f b1 = *(const v2f*)(W1 + h);
    v2f b2 = *(const v2f*)(W2 + h);
    v2f b3 = *(const v2f*)(W3 + h);
    acc0 = __builtin_amdgcn_wmma_f32_16x16x4_f32(false, a, false, b0, (short)0, acc0, false, false);
    acc1 = __builtin_amdgcn_wmma_f32_16x16x4_f32(false, a, false, b1, (short)0, acc1, false, false);
    acc2 = __builtin_amdgcn_wmma_f32_16x16x4_f32(false, a, false, b2, (short)0, acc2, false, false);
    acc3 = __builtin_amdgcn_wmma_f32_16x16x4_f32(false, a, false, b3, (short)0, acc3, false, false);
  }

  // C/D layout: VGPR v <-> M = v + 8*(lane>=16), N = lane&15
  const long mBase = (long)mtile * 16 + (lane >> 4) * 8;
  float* Cp = C + mBase * RPAD + row15;
#pragma unroll
  for (int v = 0; v < 8; v++) {
    Cp[(long)v * RPAD +  0] = acc0[v];
    Cp[(long)v * RPAD + 16] = acc1[v];
    Cp[(long)v * RPAD + 32] = acc2[v];
    Cp[(long)v * RPAD + 48] = acc3[v];
  }
}

// ---------------------------------------------------------------------------
// Stage 3: one wave (= one 32-thread block) per (b, t-tile).
// z[t,k] -> softmax over k -> alpha in LDS -> out = alpha @ X[b].
// ---------------------------------------------------------------------------
__global__ __launch_bounds__(32)
void attn_kernel(const float* __restrict__ X,   // [B][49][H]
                 const float* __restrict__ cx,  // [B*49][64]
                 const float* __restrict__ ch,  // [B*T][64]
                 const float* __restrict__ Wa,  // [49]
                 float* __restrict__ out) {     // [B][T][H]
  __shared__ float alphas[16][52];              // per-wave, K padded to 52

  const int b     = blockIdx.x >> 3;
  const int ttile = blockIdx.x & 7;
  const int lane  = threadIdx.x & 31;
  const int tloc  = lane & 15;
  const int half  = lane >> 4;
  const int t0    = ttile * 16;
  const int t     = t0 + tloc;

  const float* chp = ch + (long)(b * TDIM + t) * RPAD;

  // ---- z over k = 2*kk + half (covers k = 0..48; k==49 slot is masked) ----
  // Clamp k for addressing; select -inf afterwards. No EXEC divergence.
  float zv[25];
  float zmax = -3.0e38f;
#pragma unroll 1
  for (int kk = 0; kk < 25; kk++) {
    const int k  = kk * 2 + half;
    const int kc = imin(k, KREG - 1);
    const float* cxp = cx + (long)(b * KREG + kc) * RPAD;
    float s = 0.f;
#pragma unroll 7
    for (int r = 0; r < KREG; r++) {
      s += fast_tanh(cxp[r] + chp[r]) * Wa[r];
    }
    const float z = (k < KREG) ? s : -3.0e38f;
    zmax = fmaxf(zmax, z);
    zv[kk] = z;
  }
  // combine the two half-wave lanes that share the same t
  zmax = fmaxf(zmax, __shfl_xor(zmax, 16, 32));

  float ssum = 0.f;
#pragma unroll
  for (int kk = 0; kk < 25; kk++) {
    const int k = kk * 2 + half;
    const float e = (k < KREG) ? __expf(zv[kk] - zmax) : 0.f;
    zv[kk] = e;
    ssum += e;
  }
  ssum += __shfl_xor(ssum, 16, 32);
  const float inv = 1.0f / ssum;

#pragma unroll
  for (int kk = 0; kk < 25; kk++) {
    const int k = kk * 2 + half;            // 0..49: k==49 stores exactly 0
    alphas[tloc][k] = zv[kk] * inv;
  }
  if (half == 0) {                          // zero the K padding 50,51
    alphas[tloc][50] = 0.f;
    alphas[tloc][51] = 0.f;
  }
  // Same-wave LDS write->read is in-order (DScnt); no barrier needed in a
  // single-wave workgroup.

  // ---- context GEMM: out[16,1024] = alpha[16,52] @ X[b][52,1024] ----
  // Prebuild all 13 A fragments (reused across the 64 column tiles).
  v2f afrag[13];
#pragma unroll
  for (int kk = 0; kk < 13; kk++) {
    const int kb = kk * 4 + half * 2;
    afrag[kk].x = alphas[tloc][kb + 0];
    afrag[kk].y = alphas[tloc][kb + 1];
  }

  const float* Xb = X + (long)b * KREG * HDIM;
  float* outp = out + ((long)b * TDIM + t0 + half * 8) * HDIM + tloc;

  for (int nt = 0; nt < 64; nt++) {
    const int n0 = nt * 16;
    v8f acc = {};
#pragma unroll
    for (int kk = 0; kk < 13; kk++) {
      const int kb = kk * 4 + half * 2;
      // Clamp padded rows to 48: alpha is exactly 0 there, so the clamped
      // B values contribute nothing; clamped index is loop-invariant in nt.
      v2f bf;
      bf.x = Xb[(long)imin(kb,     KREG - 1) * HDIM + n0 + tloc];
      bf.y = Xb[(long)imin(kb + 1, KREG - 1) * HDIM + n0 + tloc];
      acc = __builtin_amdgcn_wmma_f32_16x16x4_f32(false, afrag[kk], false, bf, (short)0, acc, false, false);
    }
#pragma unroll
    for (int v = 0; v < 8; v++) outp[(long)v * HDIM + n0] = acc[v];
  }
}

// ---------------------------------------------------------------------------
extern "C" void kernel_launch(void* const* d_in, const int* in_sizes, int n_in,
                              void* d_out, int out_size, void* d_ws, size_t ws_size,
                              hipStream_t stream) {
  const float* X  = (const float*)d_in[0];   // [128,49,1024]
  const float* ht = (const float*)d_in[1];   // [128,128,1024]
  const float* Wx = (const float*)d_in[2];   // [49,1024]
  const float* Wh = (const float*)d_in[3];   // [49,1024]
  const float* Wa = (const float*)d_in[4];   // [49]
  float* out = (float*)d_out;                // [128,128,1024]

  float* cx = (float*)d_ws;                  // 6272  * 64 floats (1.6 MB)
  float* ch = cx + (long)6272 * RPAD;        // 16384 * 64 floats (4.0 MB)

  // cx: M = 128*49 = 6272 rows -> 392 m-tiles (one wave each)
  proj_kernel<<<392, 32, 0, stream>>>(X, Wx, cx);
  // ch: M = 128*128 = 16384 rows -> 1024 m-tiles
  proj_kernel<<<1024, 32, 0, stream>>>(ht, Wh, ch);
  // fused z/softmax/context: one wave per (b, t-tile) -> 128*8 = 1024 waves
  attn_kernel<<<BDIM * 8, 32, 0, stream>>>(X, cx, ch, Wa, out);
}